// RaySampler_74809740362343
// MI455X (gfx1250) — compile-verified
//
#include <hip/hip_runtime.h>

#define NRAYS 65536
#define RPB 8  // rays per block: one wave32 per ray

// ---------------------------------------------------------------------------
// One wave32 per ray:
//   1) async-stage weights[ray][1..62] into LDS (CDNA5 global_load_async_to_lds)
//   2) wave-wide shfl scan -> 63-entry CDF in LDS
//   3) 128 inverse-CDF fine samples (binary search over CDF), 4/lane
//   4) rank-based merge of (sorted) coarse+fine depth lists via binary search
//   5) block-cooperative float4 (global_store_b128) stores of all outputs
// ---------------------------------------------------------------------------
__global__ __launch_bounds__(256) void RaySampler_74809740362343_kernel(
    const float* __restrict__ origins,
    const float* __restrict__ directions,
    const float* __restrict__ nearv,
    const float* __restrict__ farv,
    const float* __restrict__ weights,
    float* __restrict__ out)
{
    __shared__ float wstage[RPB][64];    // staged raw weights
    __shared__ float cdf[RPB][64];       // cdf[0..62]
    __shared__ float zf[RPB][128];       // fine depths (sorted)
    __shared__ float zm[RPB][192];       // merged sorted depths
    __shared__ float rayinfo[RPB][8];    // ox,oy,oz,dx,dy,dz per ray

    const int lane = threadIdx.x & 31;
    const int w    = threadIdx.x >> 5;
    const int ray  = blockIdx.x * RPB + w;

    // ---- CDNA5 async global->LDS staging of the 62 pdf weights ----
    {
        unsigned lds0  = (unsigned)(unsigned long long)&wstage[w][1 + lane];
        unsigned voff0 = (unsigned)((ray * 64 + 1 + lane) * 4);
        asm volatile("global_load_async_to_lds_b32 %0, %1, %2"
                     :: "v"(lds0), "v"(voff0), "s"(weights) : "memory");
        if (lane < 30) {
            unsigned lds1  = (unsigned)(unsigned long long)&wstage[w][33 + lane];
            unsigned voff1 = (unsigned)((ray * 64 + 33 + lane) * 4);
            asm volatile("global_load_async_to_lds_b32 %0, %1, %2"
                         :: "v"(lds1), "v"(voff1), "s"(weights) : "memory");
        }
    }

    // overlap the async copy with uniform per-ray geometry loads + normalize
    __builtin_prefetch(&origins[(size_t)ray * 3], 0, 3);
    float ox = origins[ray*3+0], oy = origins[ray*3+1], oz0 = origins[ray*3+2];
    float dx = directions[ray*3+0], dy = directions[ray*3+1], dz = directions[ray*3+2];
    float rn = 1.0f / sqrtf(dx*dx + dy*dy + dz*dz);
    dx *= rn; dy *= rn; dz *= rn;
    const float zn = nearv[ray];
    const float dr = farv[ray] - zn;

    if (lane == 0) {
        rayinfo[w][0] = ox; rayinfo[w][1] = oy; rayinfo[w][2] = oz0;
        rayinfo[w][3] = dx; rayinfo[w][4] = dy; rayinfo[w][5] = dz;
    }

    asm volatile("s_wait_asynccnt 0" ::: "memory");

    // pdf elements e=0..61 map to weights[1+e]; two per lane
    float v1 = wstage[w][1 + lane] + 1e-5f;                          // e = lane
    float v2 = (lane < 30) ? (wstage[w][33 + lane] + 1e-5f) : 0.0f;  // e = 32+lane
    float s1 = v1, s2 = v2;
    #pragma unroll
    for (int d = 1; d < 32; d <<= 1) {   // wave32 inclusive scans
        float t1 = __shfl_up(s1, d, 32);
        float t2 = __shfl_up(s2, d, 32);
        if (lane >= d) { s1 += t1; s2 += t2; }
    }
    const float tot1 = __shfl(s1, 31, 32);
    const float tot  = tot1 + __shfl(s2, 31, 32);
    const float rS   = 1.0f / tot;

    if (lane == 0) cdf[w][0] = 0.0f;
    cdf[w][1 + lane] = s1 * rS;                          // cdf[1..32]
    if (lane < 30) cdf[w][33 + lane] = (tot1 + s2) * rS; // cdf[33..62]
    __syncthreads();

    // ---- 128 fine samples: inverse-CDF with searchsorted(side='right') ----
    float zfr[4];
    #pragma unroll
    for (int it = 0; it < 4; ++it) {
        int s = it * 32 + lane;
        float u = (float)s / 127.0f;
        int lo = 0, hi = 63;                       // first idx with cdf[idx] > u
        while (lo < hi) { int m = (lo + hi) >> 1; if (cdf[w][m] <= u) lo = m + 1; else hi = m; }
        int below = lo - 1; if (below < 0) below = 0;
        int above = lo;     if (above > 62) above = 62;
        float cb = cdf[w][below], ca = cdf[w][above];
        float bb = fmaf(dr, ((float)below + 0.5f) * (1.0f / 63.0f), zn);  // mids
        float ba = fmaf(dr, ((float)above + 0.5f) * (1.0f / 63.0f), zn);
        float den = ca - cb; if (den < 1e-5f) den = 1.0f;
        float t = (u - cb) / den;
        float zv = fmaf(t, ba - bb, bb);
        zf[w][s] = zv;
        zfr[it] = zv;
    }
    __syncthreads();

    // ---- rank-based merge (both lists sorted; ties: coarse before fine) ----
    #pragma unroll
    for (int it = 0; it < 4; ++it) {   // fine j: rank = j + #{coarse <= v}
        int j = it * 32 + lane;
        float v = zfr[it];
        int lo = 0, hi = 64;
        while (lo < hi) {
            int m = (lo + hi) >> 1;
            float zc = fmaf(dr, (float)m * (1.0f / 63.0f), zn);
            if (zc <= v) lo = m + 1; else hi = m;
        }
        zm[w][j + lo] = v;
    }
    #pragma unroll
    for (int it = 0; it < 2; ++it) {   // coarse i: rank = i + #{fine < v}
        int i = it * 32 + lane;
        float v = fmaf(dr, (float)i * (1.0f / 63.0f), zn);
        int lo = 0, hi = 128;
        while (lo < hi) { int m = (lo + hi) >> 1; if (zf[w][m] < v) lo = m + 1; else hi = m; }
        zm[w][i + lo] = v;
    }
    __syncthreads();

    // ---- block-cooperative wide (b128) output stores ----
    const int tid = threadIdx.x;
    const size_t blk576 = (size_t)blockIdx.x * (RPB * 576);
    const size_t blk192 = (size_t)blockIdx.x * (RPB * 192);
    float4* posq = (float4*)(out + blk576);
    float4* vdq  = (float4*)(out + (size_t)NRAYS * 576 + blk576);
    float4* zq   = (float4*)(out + (size_t)NRAYS * 1152 + blk192);
    float4* dq   = (float4*)(out + (size_t)NRAYS * 1344 + blk192);

    // positions + view_dirs: RPB*576 = 4608 floats = 1152 quads per block
    for (int q = tid; q < (RPB * 576) / 4; q += 256) {
        int f0 = q * 4;
        int rl = (int)((unsigned)f0 / 576u);   // 576 % 4 == 0 -> quad within one ray
        int rem0 = f0 - rl * 576;
        float pv[4], vv[4];
        #pragma unroll
        for (int j = 0; j < 4; ++j) {
            int rem = rem0 + j;
            int smp = (int)((unsigned)rem / 3u);
            int c = rem - smp * 3;
            float dc = rayinfo[rl][3 + c];
            float oc = rayinfo[rl][c];
            pv[j] = fmaf(zm[rl][smp], dc, oc);
            vv[j] = dc;
        }
        posq[q] = make_float4(pv[0], pv[1], pv[2], pv[3]);
        vdq[q]  = make_float4(vv[0], vv[1], vv[2], vv[3]);
    }

    // z + deltas: RPB*192 = 1536 floats = 384 quads per block
    for (int q = tid; q < (RPB * 192) / 4; q += 256) {
        int f0 = q * 4;
        int rl = (int)((unsigned)f0 / 192u);   // 192 % 4 == 0 -> quad within one ray
        int k0 = f0 - rl * 192;
        float z4[4];
        #pragma unroll
        for (int j = 0; j < 4; ++j) z4[j] = zm[rl][k0 + j];
        int kn = k0 + 4; if (kn > 191) kn = 191;       // only consumed when k0 < 188
        float znext = zm[rl][kn];
        float d4[4];
        d4[0] = z4[1] - z4[0];
        d4[1] = z4[2] - z4[1];
        d4[2] = z4[3] - z4[2];
        // last quad of a ray (k0==188): deltas[191] = z[191]-z[190] = z4[3]-z4[2]
        d4[3] = (k0 + 3 < 191) ? (znext - z4[3]) : (z4[3] - z4[2]);
        zq[q] = make_float4(z4[0], z4[1], z4[2], z4[3]);
        dq[q] = make_float4(d4[0], d4[1], d4[2], d4[3]);
    }
}

extern "C" void kernel_launch(void* const* d_in, const int* in_sizes, int n_in,
                              void* d_out, int out_size, void* d_ws, size_t ws_size,
                              hipStream_t stream) {
    (void)in_sizes; (void)n_in; (void)d_ws; (void)ws_size; (void)out_size;
    const float* origins    = (const float*)d_in[0];
    const float* directions = (const float*)d_in[1];
    const float* nearv      = (const float*)d_in[2];
    const float* farv       = (const float*)d_in[3];
    const float* weights    = (const float*)d_in[4];
    float* out = (float*)d_out;
    dim3 grid(NRAYS / RPB), block(256);
    RaySampler_74809740362343_kernel<<<grid, block, 0, stream>>>(
        origins, directions, nearv, farv, weights, out);
}